// Model_6038724018386
// MI455X (gfx1250) — compile-verified
//
#include <hip/hip_runtime.h>

// Problem constants (match reference)
#define B_    16
#define L_    256
#define A_    128
#define D_    4
#define H_    64
#define DFF_  256
#define PRED_ 64
#define COUT_ 8
#define EPS_    1e-6f
#define SELF_W_ 1e6f

typedef float v2f __attribute__((ext_vector_type(2)));
typedef float v8f __attribute__((ext_vector_type(8)));

// Fast HW approximations (v_sqrt_f32 / v_rcp_f32 / v_rsq_f32, ~1 ulp)
__device__ __forceinline__ float fast_sqrt(float x) { return __builtin_amdgcn_sqrtf(x); }
__device__ __forceinline__ float fast_rcp (float x) { return __builtin_amdgcn_rcpf(x); }
__device__ __forceinline__ float fast_rsq (float x) { return __builtin_amdgcn_rsqf(x); }

// ---------------------------------------------------------------------------
// Unnormalized adjacency element w_ij = 1/(||x_i - x_j|| + eps), diag = 1e6.
// Branchless: keeps EXEC all-1s in the WMMA inner loop.
// ---------------------------------------------------------------------------
__device__ __forceinline__ float adj_w(int i, int j,
                                       const float* __restrict__ xr,
                                       const float* __restrict__ sq) {
    const float* xi = xr + i * D_;
    const float* xj = xr + j * D_;
    float dot = xi[0] * xj[0] + xi[1] * xj[1] + xi[2] * xj[2] + xi[3] * xj[3];
    float d2  = fmaxf(sq[i] + sq[j] - 2.0f * dot, 0.0f);
    float w   = fast_rcp(fast_sqrt(d2) + EPS_);
    return (i == j) ? SELF_W_ : w;
}

__device__ __forceinline__ float fast_sigmoid(float x) {
    return fast_rcp(1.0f + __expf(-x));
}
__device__ __forceinline__ float fast_tanh(float x) {
    float e = __expf(-2.0f * x);
    return (1.0f - e) * fast_rcp(1.0f + e);
}

// ---------------------------------------------------------------------------
// Workspace zeroing (harness poisons d_ws with 0xAA)
// ---------------------------------------------------------------------------
__global__ void zero_ws_kernel(float* __restrict__ ws, int n) {
    int i = blockIdx.x * blockDim.x + threadIdx.x;
    if (i < n) ws[i] = 0.0f;
}

// ---------------------------------------------------------------------------
// Main TGCN cell: one workgroup per (b, l); 256 threads = 8 waves (wave32).
// Wave w owns output row-tile w (rows 16w..16w+15), all 4 column tiles.
// ---------------------------------------------------------------------------
__global__ __launch_bounds__(256)
void tgcn_cell_kernel(const float* __restrict__ x,
                      const float* __restrict__ Wz, const float* __restrict__ bz,
                      const float* __restrict__ Wh, const float* __restrict__ bh,
                      const float* __restrict__ Lzw, const float* __restrict__ Lzb,
                      const float* __restrict__ Lhw, const float* __restrict__ Lhb,
                      float* __restrict__ pooled_acc) {
    __shared__ float xr[A_ * D_];        // 128x4 agent features
    __shared__ float sq[A_];             // per-agent squared norm
    __shared__ float rowsum[A_];         // adjacency row sums
    __shared__ float dinv[A_];           // D^{-1/2}
    __shared__ float buf[A_][H_ + 2];    // y = dinv*(X@W) then c (padded)
    __shared__ float Lbuf[H_][H_ + 2];   // gate linear (first 64 rows of [2H,H])
    __shared__ float Wg[D_ * H_];        // GCN conv weight [4][64]
    __shared__ float bg[H_];             // GCN conv bias
    __shared__ float lb[H_];             // gate linear bias
    __shared__ float colsum[H_];         // per-column (feature) sum over agents

    const int tid  = threadIdx.x;
    const int lane = tid & 31;
    const int wv   = tid >> 5;           // wave id = row tile
    const int bidx = blockIdx.x;         // b*L + l
    const int b    = bidx >> 8;          // L_ == 256

    // --- load xr (512 floats via 256 threads) ---
    const float* xg = x + (size_t)bidx * (A_ * D_);
    xr[2 * tid]     = xg[2 * tid];
    xr[2 * tid + 1] = xg[2 * tid + 1];
    if (tid < H_) colsum[tid] = 0.0f;
    if (tid < A_) rowsum[tid] = 0.0f;
    __syncthreads();

    if (tid < A_) {
        float s = 0.0f;
#pragma unroll
        for (int d = 0; d < D_; ++d) { float v = xr[tid * D_ + d]; s += v * v; }
        sq[tid] = s;
    }
    __syncthreads();

    // --- adjacency row sums: 2 threads per row ---
    {
        int row  = tid & (A_ - 1);
        int part = tid >> 7;
        float s = 0.0f;
        for (int j = part * 64; j < part * 64 + 64; ++j)
            s += adj_w(row, j, xr, sq);
        atomicAdd(&rowsum[row], s);
    }
    __syncthreads();
    if (tid < A_) dinv[tid] = fast_rsq(rowsum[tid]);
    __syncthreads();

    const int mrow  = 16 * wv + (lane & 15);     // A-fragment row (this wave)
    const int khalf = (lane >> 4) << 1;          // K sub-offset per lane half

    v8f zf[4];                                   // Z fragments (gate 0 -> gate 1)

    for (int g = 0; g < 2; ++g) {
        const float* Wp  = g ? Wh  : Wz;
        const float* bp  = g ? bh  : bz;
        const float* Lwp = g ? Lhw : Lzw;
        const float* Lbp = g ? Lhb : Lzb;

        // --- stage per-gate weights into LDS ---
        if (tid < D_ * H_) Wg[tid] = Wp[tid];
        if (tid < H_) { bg[tid] = bp[tid]; lb[tid] = Lbp[tid]; }
        for (int e = tid; e < H_ * H_; e += 256)        // rows 0..63 of [2H,H]
            Lbuf[e >> 6][e & 63] = Lwp[e];
        __syncthreads();

        // --- y[j][h] = dinv[j] * sum_d xr[j][d] * Wg[d][h] ---
        for (int e = tid; e < A_ * H_; e += 256) {
            int j = e >> 6, h = e & 63;
            float s = 0.0f;
#pragma unroll
            for (int d = 0; d < D_; ++d) s += xr[j * D_ + d] * Wg[d * H_ + h];
            buf[j][h] = dinv[j] * s;
        }
        __syncthreads();

        // --- GEMM1: acc = Wadj(row-tile wv) @ y   (M=16,N=64,K=128) ---
        v8f acc[4] = {};
        for (int ks = 0; ks < 32; ++ks) {
            int k0 = 4 * ks + khalf;
            v2f a;
            a.x = adj_w(mrow, k0,     xr, sq);
            a.y = adj_w(mrow, k0 + 1, xr, sq);
#pragma unroll
            for (int ct = 0; ct < 4; ++ct) {
                int n = 16 * ct + (lane & 15);
                v2f bb;
                bb.x = buf[k0][n];
                bb.y = buf[k0 + 1][n];
                acc[ct] = __builtin_amdgcn_wmma_f32_16x16x4_f32(
                    false, a, false, bb, (short)0, acc[ct], false, false);
            }
        }
        __syncthreads();    // all y reads complete before overwrite

        // --- epilogue: c = dinv_row * acc + bias, staged back into buf ---
        {
            int mbase = 16 * wv + ((lane >> 4) << 3);
#pragma unroll
            for (int ct = 0; ct < 4; ++ct) {
                int n = 16 * ct + (lane & 15);
#pragma unroll
                for (int t = 0; t < 8; ++t)
                    buf[mbase + t][n] = dinv[mbase + t] * acc[ct][t] + bg[n];
            }
        }
        __syncthreads();

        // --- GEMM2: t = c @ Lg   (M=16,N=64,K=64) ---
        v8f acc2[4] = {};
        for (int ks = 0; ks < 16; ++ks) {
            int k0 = 4 * ks + khalf;
            v2f a;
            a.x = buf[mrow][k0];
            a.y = buf[mrow][k0 + 1];
#pragma unroll
            for (int ct = 0; ct < 4; ++ct) {
                int n = 16 * ct + (lane & 15);
                v2f bb;
                bb.x = Lbuf[k0][n];
                bb.y = Lbuf[k0 + 1][n];
                acc2[ct] = __builtin_amdgcn_wmma_f32_16x16x4_f32(
                    false, a, false, bb, (short)0, acc2[ct], false, false);
            }
        }

        if (g == 0) {
            // Z = sigmoid(t_z + lb)
#pragma unroll
            for (int ct = 0; ct < 4; ++ct) {
                int n = 16 * ct + (lane & 15);
#pragma unroll
                for (int t = 0; t < 8; ++t)
                    zf[ct][t] = fast_sigmoid(acc2[ct][t] + lb[n]);
            }
        } else {
            // h = relu((1-Z) * tanh(t_h + lb)); accumulate column sums
#pragma unroll
            for (int ct = 0; ct < 4; ++ct) {
                int n = 16 * ct + (lane & 15);
                float s = 0.0f;
#pragma unroll
                for (int t = 0; t < 8; ++t) {
                    float ht = fast_tanh(acc2[ct][t] + lb[n]);
                    float hv = fmaxf((1.0f - zf[ct][t]) * ht, 0.0f);
                    s += hv;
                }
                atomicAdd(&colsum[n], s);
            }
        }
        __syncthreads();    // protects Lbuf/buf reuse and finalizes colsum
    }

    // --- accumulate into global per-batch pool (sum over agents & seq) ---
    if (tid < H_) atomicAdd(&pooled_acc[b * H_ + tid], colsum[tid]);
}

// ---------------------------------------------------------------------------
// Decoder: pooled -> relu(pooled@D1 + b1) @ D2 + b2, one block per batch
// ---------------------------------------------------------------------------
__global__ __launch_bounds__(256)
void decoder_kernel(const float* __restrict__ acc,
                    const float* __restrict__ D1w, const float* __restrict__ D1b,
                    const float* __restrict__ D2w, const float* __restrict__ D2b,
                    float* __restrict__ out) {
    __shared__ float pooled[H_];
    __shared__ float d1[DFF_];
    const int b = blockIdx.x, tid = threadIdx.x;

    if (tid < H_) pooled[tid] = acc[b * H_ + tid] * (1.0f / (float)(A_ * L_));
    __syncthreads();

    {
        float s = D1b[tid];
        for (int k = 0; k < H_; ++k) s += pooled[k] * D1w[k * DFF_ + tid];
        d1[tid] = fmaxf(s, 0.0f);
    }
    __syncthreads();

    for (int o = tid; o < PRED_ * COUT_; o += 256) {
        float s = D2b[o];
        for (int k = 0; k < DFF_; ++k) s += d1[k] * D2w[k * (PRED_ * COUT_) + o];
        out[(size_t)b * (PRED_ * COUT_) + o] = s;
    }
}

// ---------------------------------------------------------------------------
extern "C" void kernel_launch(void* const* d_in, const int* in_sizes, int n_in,
                              void* d_out, int out_size, void* d_ws, size_t ws_size,
                              hipStream_t stream) {
    const float* x   = (const float*)d_in[0];
    const float* Wz  = (const float*)d_in[4];
    const float* bz  = (const float*)d_in[5];
    const float* Wh  = (const float*)d_in[8];
    const float* bh  = (const float*)d_in[9];
    const float* Lzw = (const float*)d_in[10];
    const float* Lzb = (const float*)d_in[11];
    const float* Lhw = (const float*)d_in[14];
    const float* Lhb = (const float*)d_in[15];
    const float* D1w = (const float*)d_in[16];
    const float* D1b = (const float*)d_in[17];
    const float* D2w = (const float*)d_in[18];
    const float* D2b = (const float*)d_in[19];

    float* ws  = (float*)d_ws;
    float* out = (float*)d_out;

    zero_ws_kernel<<<dim3((B_ * H_ + 255) / 256), dim3(256), 0, stream>>>(ws, B_ * H_);
    tgcn_cell_kernel<<<dim3(B_ * L_), dim3(256), 0, stream>>>(
        x, Wz, bz, Wh, bh, Lzw, Lzb, Lhw, Lhb, ws);
    decoder_kernel<<<dim3(B_), dim3(256), 0, stream>>>(ws, D1w, D1b, D2w, D2b, out);
}